// RotMat_62766652064240
// MI455X (gfx1250) — compile-verified
//
#include <hip/hip_runtime.h>
#include <cstdint>

#define N 2048
#define M 2047            // N-1 rounds, modulus of the circle method
#define HALF 1024         // pairs per round
#define COLS 8            // columns of U per workgroup
#define NTHREADS 256

typedef float f4 __attribute__((ext_vector_type(4)));
typedef float f2 __attribute__((ext_vector_type(2)));

__device__ __forceinline__ void pair_rows(int k, int r, int& i, int& j) {
    // circle-method round robin, closed form (player 0 fixed)
    int a, b;
    if (k == 0) {
        a = 0;
        b = M - r;                          // ((M-1-r) mod M) + 1, r <= M-1 so no wrap
    } else {
        a = k - 1 - r;       if (a < 0) a += M;  a += 1;
        b = (N - 2) - k - r; if (b < 0) b += M;  b += 1;
    }
    i = a < b ? a : b;
    j = a < b ? b : a;
}

__device__ __forceinline__ int theta_index(int i, int j) {
    // row-major upper-triangular index for pair (i,j), i<j
    return i * N - ((i * (i + 1)) >> 1) + (j - i - 1);
}

extern "C" __global__ __launch_bounds__(NTHREADS)
void rotmat_kernel(const float* __restrict__ thetas, float* __restrict__ out) {
    extern __shared__ float smem[];
    float* Ub  = smem;                      // N * COLS floats, row-major 2048 x 8, NO pad
    float* tb  = smem + N * COLS;           // 2 * HALF floats  (theta ping-pong)
    float* csb = tb + 2 * HALF;             // 2 * 2*HALF floats (cos/sin ping-pong)

    const int t     = threadIdx.x;
    const int wg    = blockIdx.x;
    const int w     = t >> 5;               // wave id (0..7), wave32
    const int l     = t & 31;
    const int cq    = t & 1;                // column quad: cols 4cq..4cq+3
    const int g     = t >> 1;               // pair-group (0..127), 8 pairs each
    const int gcol0 = wg * COLS;
    const uint64_t thbase = (uint64_t)(uintptr_t)thetas;

    // ---- init U block: identity columns (row-major) ----
    for (int idx = t; idx < N * COLS; idx += NTHREADS) {
        int c   = idx & (COLS - 1);
        int row = idx >> 3;
        Ub[idx] = (row == gcol0 + c) ? 1.0f : 0.0f;
    }

    // ---- stage round 0: async gather thetas -> LDS, then sincos -> cs buffer ----
    {
        float* tdst = tb;                   // parity 0
        for (int q = 0; q < 4; ++q) {
            int k = w * 128 + q * 32 + l;   // wave-private pair range [128w, 128w+127]
            int i, j; pair_rows(k, 0, i, j);
            uint32_t goff = (uint32_t)theta_index(i, j) * 4u;
            uint32_t loff = (uint32_t)(uintptr_t)(tdst + k);
            asm volatile("global_load_async_to_lds_b32 %0, %1, %2"
                         :: "v"(loff), "v"(goff), "s"(thbase) : "memory");
        }
        asm volatile("s_wait_asynccnt 0x0" ::: "memory");
        for (int q = 0; q < 4; ++q) {
            int k = w * 128 + q * 32 + l;
            float sv, cv;
            sincosf(tdst[k], &sv, &cv);
            csb[2 * k]     = cv;
            csb[2 * k + 1] = sv;
        }
    }
    __syncthreads();

    // ---- 2047 rounds, software-pipelined staging ----
    for (int r = 0; r < M; ++r) {
        const int p = r & 1;

        // issue async gather for round r+1 early (hidden behind rotate)
        if (r + 1 < M) {
            float* tdst = tb + (p ^ 1) * HALF;
            for (int q = 0; q < 4; ++q) {
                int k = w * 128 + q * 32 + l;
                int i, j; pair_rows(k, r + 1, i, j);
                uint32_t goff = (uint32_t)theta_index(i, j) * 4u;
                uint32_t loff = (uint32_t)(uintptr_t)(tdst + k);
                asm volatile("global_load_async_to_lds_b32 %0, %1, %2"
                             :: "v"(loff), "v"(goff), "s"(thbase) : "memory");
            }
        }

        // rotate round r: this thread applies 8 pairs to its 4-column slice (b128 DS ops)
        {
            f4* U4 = (f4*)Ub;                         // row r -> U4[2r + cq]
            const f2* cs2 = (const f2*)(csb + p * (2 * HALF));
            for (int q = 0; q < 8; ++q) {
                int k = g * 8 + q;
                int i, j; pair_rows(k, r, i, j);
                f2 cs = cs2[k];
                float cv = cs.x, sv = cs.y;
                f4 x = U4[2 * i + cq];
                f4 y = U4[2 * j + cq];
                U4[2 * i + cq] = cv * x - sv * y;
                U4[2 * j + cq] = sv * x + cv * y;
            }
        }

        // finish staging round r+1 (wave-private: no barrier needed for c/s)
        if (r + 1 < M) {
            const float* tsrc = tb + (p ^ 1) * HALF;
            float* csn = csb + (p ^ 1) * (2 * HALF);
            asm volatile("s_wait_asynccnt 0x0" ::: "memory");
            for (int q = 0; q < 4; ++q) {
                int k = w * 128 + q * 32 + l;
                float sv, cv;
                sincosf(tsrc[k], &sv, &cv);
                csn[2 * k]     = cv;
                csn[2 * k + 1] = sv;
            }
        }
        __syncthreads();   // rows of U mix across waves between rounds
    }

    // ---- epilogue 1: TDM store of the contiguous 2048x8 LDS tile -> strided rows of out ----
#if __has_builtin(__builtin_amdgcn_tensor_store_from_lds)
    if (t == 0) {
        typedef unsigned int v4u __attribute__((ext_vector_type(4)));
        typedef int v4i __attribute__((ext_vector_type(4)));
        typedef int v8i __attribute__((ext_vector_type(8)));
        const uint32_t ldsoff = (uint32_t)(uintptr_t)Ub;
        const uint64_t gaddr  = (uint64_t)(uintptr_t)(out + gcol0);
        // D# group0: count=1 | lds_addr | global_addr[56:0] | type=2
        v4u g0 = { 1u, ldsoff, (uint32_t)gaddr,
                   (uint32_t)(gaddr >> 32) | 0x80000000u };
        // D# group1: data_size=4B; tensor_dim0=8, tensor_dim1=2048;
        //            tile_dim0=8, tile_dim1=2048, tile_dim2=0; dim0_stride=2048
        v8i g1 = { 0x00020000, 0x00080000, 0x08000000, 0x00080000,
                   0x00000800, 2048, 0, 0 };
        v4i gz4 = { 0, 0, 0, 0 };                     // 2-D tensor: groups 2/3 unused
        v8i gz8 = { 0, 0, 0, 0, 0, 0, 0, 0 };         // extended words unused (zero)
        __builtin_amdgcn_tensor_store_from_lds(g0, g1, gz4, gz4, gz8, 0);
        __builtin_amdgcn_s_wait_tensorcnt((short)0);
    }
#endif

    // ---- epilogue 2: vectorized scalar path (writes identical bytes; race-free) ----
    {
        const f4* U4 = (const f4*)Ub;
        f4* O4 = (f4*)out;
        for (int idx = t; idx < N * 2; idx += NTHREADS) {
            int c4  = idx & 1;
            int row = idx >> 1;
            O4[row * (N / 4) + wg * 2 + c4] = U4[row * 2 + c4];
        }
    }
}

extern "C" void kernel_launch(void* const* d_in, const int* in_sizes, int n_in,
                              void* d_out, int out_size, void* d_ws, size_t ws_size,
                              hipStream_t stream) {
    (void)in_sizes; (void)n_in; (void)d_ws; (void)ws_size; (void)out_size;
    const float* thetas = (const float*)d_in[0];
    float* out = (float*)d_out;

    const int smem_bytes = (N * COLS + 2 * HALF + 4 * HALF) * (int)sizeof(float); // 90112 B
    (void)hipFuncSetAttribute((const void*)rotmat_kernel,
                              hipFuncAttributeMaxDynamicSharedMemorySize, smem_bytes);
    rotmat_kernel<<<N / COLS, NTHREADS, smem_bytes, stream>>>(thetas, out);
}